// Net_10282151707399
// MI455X (gfx1250) — compile-verified
//
#include <hip/hip_runtime.h>
#include <hip/hip_bf16.h>
#include <math.h>

// ---------------------------------------------------------------------------
// CDNA5 (gfx1250) implementation. Heavy GEMMs (H@H^T, adj@X, X@W, adj@S) run
// on V_WMMA_F32_16X16X4_F32 (wave32, one 16x16 tile per wave, K stepped by 4).
// ---------------------------------------------------------------------------

typedef float v2f __attribute__((ext_vector_type(2)));
typedef float v8f __attribute__((ext_vector_type(8)));

static __device__ inline v8f wmma_f32_16x16x4(v2f a, v2f b, v8f c) {
  // 8 args: (neg_a, A, neg_b, B, c_mod, C, reuse_a, reuse_b)
  return __builtin_amdgcn_wmma_f32_16x16x4_f32(false, a, false, b, (short)0, c,
                                               false, false);
}

// ------------------------------- kernels -----------------------------------

__global__ void k_zero(float* __restrict__ p, long long n) {
  long long i = (long long)blockIdx.x * blockDim.x + threadIdx.x;
  long long st = (long long)gridDim.x * blockDim.x;
  for (; i < n; i += st) p[i] = 0.f;
}

// pack src[sr,sc] (row-major) into dst[dr,dc] zero-padded
__global__ void k_pack(const float* __restrict__ src, int sr, int sc,
                       float* __restrict__ dst, int dr, int dc) {
  int i = blockIdx.x * blockDim.x + threadIdx.x;
  if (i >= dr * dc) return;
  int r = i / dc, c = i % dc;
  dst[i] = (r < sr && c < sc) ? src[r * sc + c] : 0.f;
}

__global__ void k_copycol(const float* __restrict__ src, int ls, int rows, int w,
                          float* __restrict__ dst, int ld, int off) {
  int i = blockIdx.x * blockDim.x + threadIdx.x;
  if (i >= rows * w) return;
  int r = i / w, c = i % w;
  dst[(size_t)r * ld + off + c] = src[(size_t)r * ls + c];
}

// Fused per-image CNN: conv1(5x5,1->20)+relu+pool2 -> conv2(5x5,20->50)+relu
// +pool2 -> fc(800->50)+relu. One workgroup per image, all staged in LDS.
__global__ __launch_bounds__(256) void k_cnn(
    const float* __restrict__ x, const float* __restrict__ w1,
    const float* __restrict__ b1, const float* __restrict__ w2,
    const float* __restrict__ b2, const float* __restrict__ fw,
    const float* __restrict__ fb, float* __restrict__ H,
    float* __restrict__ sq) {
  __shared__ float im[784];
  __shared__ float p1[2880];  // 20x12x12
  __shared__ float p2[800];   // 50x4x4
  __shared__ float cw1[500];
  __shared__ float hh[64];
  int t = threadIdx.x, img = blockIdx.x;
  for (int i = t; i < 784; i += 256) im[i] = x[img * 784 + i];
  for (int i = t; i < 500; i += 256) cw1[i] = w1[i];
  __syncthreads();
  for (int o = t; o < 2880; o += 256) {
    int c = o / 144, rem = o % 144, py = rem / 12, px = rem % 12;
    float m = -1e30f;
    for (int dy = 0; dy < 2; ++dy)
      for (int dx = 0; dx < 2; ++dx) {
        int oy = py * 2 + dy, ox = px * 2 + dx;
        float s = b1[c];
        #pragma unroll
        for (int ky = 0; ky < 5; ++ky)
          #pragma unroll
          for (int kx = 0; kx < 5; ++kx)
            s += im[(oy + ky) * 28 + ox + kx] * cw1[c * 25 + ky * 5 + kx];
        m = fmaxf(m, s);
      }
    p1[o] = fmaxf(m, 0.f);  // relu(max) == max(relu)
  }
  __syncthreads();
  for (int o = t; o < 800; o += 256) {
    int c = o / 16, rem = o % 16, py = rem / 4, px = rem % 4;
    float m = -1e30f;
    for (int dy = 0; dy < 2; ++dy)
      for (int dx = 0; dx < 2; ++dx) {
        int oy = py * 2 + dy, ox = px * 2 + dx;
        float s = b2[c];
        for (int ic = 0; ic < 20; ++ic) {
          const float* wp = w2 + (c * 20 + ic) * 25;
          const float* ip = p1 + ic * 144 + oy * 12 + ox;
          #pragma unroll
          for (int ky = 0; ky < 5; ++ky)
            #pragma unroll
            for (int kx = 0; kx < 5; ++kx) s += ip[ky * 12 + kx] * wp[ky * 5 + kx];
        }
        m = fmaxf(m, s);
      }
    p2[o] = fmaxf(m, 0.f);
  }
  __syncthreads();
  if (t < 64) hh[t] = 0.f;
  __syncthreads();
  if (t < 50) {
    float s = fb[t];
    for (int i = 0; i < 800; ++i) s += p2[i] * fw[i * 50 + t];
    hh[t] = fmaxf(s, 0.f);
  }
  __syncthreads();
  if (t < 64) H[(size_t)img * 64 + t] = hh[t];  // zero-padded cols 50..63
  if (t == 0) {
    float s = 0.f;
    for (int i = 0; i < 50; ++i) s += hh[i] * hh[i];
    sq[img] = s;
  }
}

// D2 = sq_i - 2*H H^T + sq_j, clipped >= 0. One 16x16 tile per wave (WMMA f32).
__global__ __launch_bounds__(256) void k_gram_d2(const float* __restrict__ H,
                                                 const float* __restrict__ sq,
                                                 float* __restrict__ D2, int Nn) {
  int wav = blockIdx.x * (blockDim.x >> 5) + (threadIdx.x >> 5);
  int lane = threadIdx.x & 31;
  int tiles = Nn >> 4;
  if (wav >= tiles * tiles) return;  // wave-uniform
  int tM = (wav / tiles) << 4, tN = (wav % tiles) << 4;
  int hf = lane >> 4, idx = lane & 15;
  const float* ra = H + (size_t)(tM + idx) * 64 + hf * 2;
  const float* rb = H + (size_t)(tN + idx) * 64 + hf * 2;  // B = H^T
  v8f c = {0.f, 0.f, 0.f, 0.f, 0.f, 0.f, 0.f, 0.f};
  #pragma unroll
  for (int k = 0; k < 64; k += 4) {
    v2f a = *(const v2f*)(ra + k);
    v2f b = *(const v2f*)(rb + k);
    c = wmma_f32_16x16x4(a, b, c);
  }
  int col = tN + idx;
  #pragma unroll
  for (int i = 0; i < 8; ++i) {
    int r = tM + i + hf * 8;
    float d2 = sq[r] - 2.f * c[i] + sq[col];
    D2[(size_t)r * Nn + col] = fmaxf(d2, 0.f);
  }
}

// histogram of first-passing-threshold index k (thr_k = 3.5 + 0.5k)
__global__ void k_edge_hist(const float* __restrict__ D2, int Nn,
                            unsigned int* __restrict__ hist) {
  __shared__ unsigned int h[80];
  int t = threadIdx.x;
  for (int i = t; i < 80; i += blockDim.x) h[i] = 0u;
  __syncthreads();
  long long total = (long long)Nn * Nn;
  long long i = (long long)blockIdx.x * blockDim.x + t;
  long long st = (long long)gridDim.x * blockDim.x;
  for (; i < total; i += st) {
    int r = (int)(i / Nn), cc = (int)(i % Nn);
    if (r == cc) continue;
    float s = sqrtf(D2[i]);
    int k = (int)floorf((s - 3.5f) * 2.f) + 1;  // smallest k with s < thr_k
    if (k < 0) k = 0;
    if (k > 64) k = 64;
    atomicAdd(&h[k], 1u);
  }
  __syncthreads();
  for (int k2 = t; k2 < 80; k2 += blockDim.x)
    if (h[k2]) atomicAdd(&hist[k2], h[k2]);
}

__global__ void k_pick_thr(const unsigned int* __restrict__ hist,
                           float* __restrict__ scal, float minEdges) {
  if (threadIdx.x == 0 && blockIdx.x == 0) {
    unsigned long long cum = 0;
    int k = 64;
    for (int i = 0; i < 64; ++i) {
      cum += hist[i];
      if ((float)cum >= minEdges) { k = i; break; }
    }
    float thr = 3.5f + 0.5f * (float)k;
    scal[0] = thr * thr;
  }
}

__global__ void k_adj_build(float* __restrict__ A, int Nn,
                            const float* __restrict__ scal) {
  float thr2 = scal[0];
  long long total = (long long)Nn * Nn;
  long long i = (long long)blockIdx.x * blockDim.x + threadIdx.x;
  long long st = (long long)gridDim.x * blockDim.x;
  for (; i < total; i += st) {
    int r = (int)(i / Nn), c = (int)(i % Nn);
    A[i] = (r != c && A[i] < thr2) ? 1.f : 0.f;
  }
}

__global__ void k_invdeg(const float* __restrict__ A, int lda, int K,
                         float* __restrict__ out) {
  __shared__ float red[256];
  int r = blockIdx.x, t = threadIdx.x;
  const float* row = A + (size_t)r * lda;
  float s = 0.f;
  for (int k = t; k < K; k += blockDim.x) s += row[k];
  red[t] = s;
  __syncthreads();
  for (int w = 128; w > 0; w >>= 1) {
    if (t < w) red[t] += red[t + w];
    __syncthreads();
  }
  if (t == 0) out[r] = 1.f / fmaxf(red[0], 1.f);
}

// Generic f32 WMMA GEMM: C = rowscale .* (A@B) + addm + bias, optional relu.
// M,N multiples of 16; K multiple of 4; padded regions of operands are zero.
__global__ __launch_bounds__(256) void k_gemm_wmma(
    const float* __restrict__ A, int lda, const float* __restrict__ B, int ldb,
    float* __restrict__ C, int ldc, int M, int N, int K,
    const float* __restrict__ addm, const float* __restrict__ bias, int biasN,
    const float* __restrict__ rowscale, int relu) {
  int wav = blockIdx.x * (blockDim.x >> 5) + (threadIdx.x >> 5);
  int lane = threadIdx.x & 31;
  int tn = N >> 4;
  int tiles = (M >> 4) * tn;
  if (wav >= tiles) return;  // wave-uniform
  int tM = (wav / tn) << 4, tN = (wav % tn) << 4;
  int hf = lane >> 4, idx = lane & 15;
  const float* ra = A + (size_t)(tM + idx) * lda + hf * 2;
  const float* rb = B + (size_t)(hf * 2) * ldb + tN + idx;
  v8f c = {0.f, 0.f, 0.f, 0.f, 0.f, 0.f, 0.f, 0.f};
  for (int k = 0; k < K; k += 4) {
    v2f a = *(const v2f*)(ra + k);
    v2f b;
    b.x = rb[(size_t)k * ldb];
    b.y = rb[(size_t)(k + 1) * ldb];
    c = wmma_f32_16x16x4(a, b, c);
  }
  int col = tN + idx;
  float bi = (bias && col < biasN) ? bias[col] : 0.f;
  #pragma unroll
  for (int i = 0; i < 8; ++i) {
    int r = tM + i + hf * 8;
    float v = c[i];
    if (rowscale) v *= rowscale[r];
    if (addm) v += addm[(size_t)r * ldc + col];
    v += bi;
    if (relu) v = fmaxf(v, 0.f);
    C[(size_t)r * ldc + col] = v;
  }
}

__global__ void k_bn_stats(const float* __restrict__ Y, int ld, int Nreal,
                           float* __restrict__ stats) {
  __shared__ float s1[256], s2[256];
  int ch = blockIdx.x, t = threadIdx.x;
  float a = 0.f, b = 0.f;
  for (int r = t; r < Nreal; r += blockDim.x) {
    float v = Y[(size_t)r * ld + ch];
    a += v;
    b += v * v;
  }
  s1[t] = a;
  s2[t] = b;
  __syncthreads();
  for (int w = 128; w > 0; w >>= 1) {
    if (t < w) { s1[t] += s1[t + w]; s2[t] += s2[t + w]; }
    __syncthreads();
  }
  if (t == 0) {
    float m = s1[0] / (float)Nreal;
    float v = s2[0] / (float)Nreal - m * m;
    stats[ch] = m;
    stats[64 + ch] = fmaxf(v, 0.f);
  }
}

__global__ void k_bn_apply(float* __restrict__ Y, int ld, int Mpad,
                           const float* __restrict__ stats,
                           const float* __restrict__ g,
                           const float* __restrict__ b, int cN) {
  int i = blockIdx.x * blockDim.x + threadIdx.x;
  if (i >= Mpad * ld) return;
  int c = i % ld;
  float gg = (c < cN) ? g[c] : 0.f;
  float bb = (c < cN) ? b[c] : 0.f;
  float m = stats[c], v = stats[64 + c];
  Y[i] = (Y[i] - m) * rsqrtf(v + 1e-5f) * gg + bb;
}

__global__ void k_softmax_rows(float* __restrict__ S, int ld, int Nreal, int C) {
  int r = blockIdx.x * blockDim.x + threadIdx.x;
  if (r >= Nreal) return;
  float* row = S + (size_t)r * ld;
  float m = -1e30f;
  for (int c = 0; c < C; ++c) m = fmaxf(m, row[c]);
  float s = 0.f;
  for (int c = 0; c < C; ++c) {
    float e = __expf(row[c] - m);
    row[c] = e;
    s += e;
  }
  float inv = 1.f / s;
  for (int c = 0; c < C; ++c) row[c] *= inv;
}

// out[16,64] = S^T X over real rows (also used for adjn = S^T (A S))
__global__ void k_stx(const float* __restrict__ S, int ls,
                      const float* __restrict__ X, int lx, int Nreal,
                      float* __restrict__ out) {
  int i = blockIdx.x * blockDim.x + threadIdx.x;
  if (i >= 16 * 64) return;
  int c = i / 64, f = i % 64;
  float s = 0.f;
  for (int n = 0; n < Nreal; ++n)
    s += S[(size_t)n * ls + c] * X[(size_t)n * lx + f];
  out[i] = s;  // out stride 64, rows >=16 untouched (pre-zeroed)
}

__global__ void k_link(const float* __restrict__ A, int lda,
                       const float* __restrict__ S, int ls, int Nreal,
                       float* __restrict__ acc) {
  __shared__ float red[256];
  long long total = (long long)Nreal * Nreal;
  long long i = (long long)blockIdx.x * blockDim.x + threadIdx.x;
  long long st = (long long)gridDim.x * blockDim.x;
  float a = 0.f;
  for (; i < total; i += st) {
    int r = (int)(i / Nreal), c = (int)(i % Nreal);
    const float* sr = S + (size_t)r * ls;
    const float* sc = S + (size_t)c * ls;
    float d = 0.f;
    #pragma unroll
    for (int k = 0; k < 16; ++k) d += sr[k] * sc[k];  // padded cols are 0
    float e = A[(size_t)r * lda + c] - d;
    a += e * e;
  }
  int t = threadIdx.x;
  red[t] = a;
  __syncthreads();
  for (int w = 128; w > 0; w >>= 1) {
    if (t < w) red[t] += red[t + w];
    __syncthreads();
  }
  if (t == 0) atomicAdd(acc, red[0]);
}

__global__ void k_ent(const float* __restrict__ S, int ls, int Nreal, int C,
                      float* __restrict__ acc) {
  int r = blockIdx.x * blockDim.x + threadIdx.x;
  if (r >= Nreal) return;
  const float* row = S + (size_t)r * ls;
  float s = 0.f;
  for (int c = 0; c < C; ++c) s += -row[c] * logf(row[c] + 1e-15f);
  atomicAdd(acc, s);
}

__global__ __launch_bounds__(128) void k_final(
    const float* __restrict__ CAT, int ld, int Nreal,
    const float* __restrict__ l1W, const float* __restrict__ l1b,
    const float* __restrict__ l2W, const float* __restrict__ l2b,
    const float* __restrict__ scal, float* __restrict__ out) {
  __shared__ float v96[96];
  __shared__ float h32[32];
  __shared__ float lg[2];
  int t = threadIdx.x;
  if (t < 96) {
    float s = 0.f;
    for (int n = 0; n < Nreal; ++n) s += CAT[(size_t)n * ld + t];
    v96[t] = s / (float)Nreal;
  }
  __syncthreads();
  if (t < 32) {
    float s = l1b[t];
    for (int i = 0; i < 96; ++i) s += v96[i] * l1W[i * 32 + t];
    h32[t] = fmaxf(s, 0.f);
  }
  __syncthreads();
  if (t < 2) {
    float s = l2b[t];
    for (int i = 0; i < 32; ++i) s += h32[i] * l2W[i * 2 + t];
    lg[t] = s;
  }
  __syncthreads();
  if (t == 0) {
    float m = fmaxf(lg[0], lg[1]);
    float lz = m + logf(expf(lg[0] - m) + expf(lg[1] - m));
    out[0] = lg[0] - lz;
    out[1] = lg[1] - lz;
    out[2] = sqrtf(scal[1]) / (4096.f * 4096.f) + sqrtf(scal[2]) / 49.f;
    out[3] = scal[3] / 4096.f + scal[4] / 7.f;
  }
}

// ----------------------------- host-side plumbing ---------------------------

namespace {

constexpr size_t SZ_ADJ1 = 4096ull * 4096;
constexpr size_t OFF_ADJ1 = 0;
constexpr size_t OFF_H = OFF_ADJ1 + SZ_ADJ1;        // 4096x64
constexpr size_t OFF_SQ = OFF_H + 4096ull * 64;     // 4096
constexpr size_t OFF_IDG1 = OFF_SQ + 4096;          // 4096
constexpr size_t OFF_IDG2 = OFF_IDG1 + 4096;        // 64
constexpr size_t OFF_IDG3 = OFF_IDG2 + 64;          // 64
constexpr size_t OFF_HIST = OFF_IDG3 + 64;          // 128 (uint)
constexpr size_t OFF_SCAL = OFF_HIST + 128;         // [0]=thr2 [1]=l1 [2]=l2 [3]=e1 [4]=e2
constexpr size_t OFF_STAT = OFF_SCAL + 64;          // 128 (mean|var)
constexpr size_t OFF_WSL = OFF_STAT + 128;          // 40 slots x 8192
constexpr size_t OFF_XB = OFF_WSL + 40ull * 8192;   // 14 buffers x 4096x64
constexpr size_t OFF_CATA = OFF_XB + 14ull * 262144;
constexpr size_t OFF_CATB = OFF_CATA + 4096ull * 128;
constexpr size_t OFF_END = OFF_CATB + 4096ull * 128;

struct Ctx {
  float* W;
  hipStream_t st;
  float* slot(int i) const { return W + OFF_WSL + (size_t)i * 8192; }
  float* xb(int i) const { return W + OFF_XB + (size_t)i * 262144; }
};

struct GnnP {
  const float *Wr[3], *Wx[3], *bias[3], *g[3], *b[3];
  int cin, hid, cout;
  const float *linW, *linb;
  int lincin, lincout;
};

GnnP mk_gnn(void* const* din, int base, int cin, int hid, int cout, int lincin) {
  auto F = [&](int i) { return (const float*)din[i]; };
  GnnP p;
  p.b[0] = F(base + 0);  p.g[0] = F(base + 1);
  p.b[1] = F(base + 2);  p.g[1] = F(base + 3);
  p.b[2] = F(base + 4);  p.g[2] = F(base + 5);
  p.Wr[0] = F(base + 6); p.Wx[0] = F(base + 7); p.bias[0] = F(base + 8);
  p.Wr[1] = F(base + 9); p.Wx[1] = F(base + 10); p.bias[1] = F(base + 11);
  p.Wr[2] = F(base + 12); p.Wx[2] = F(base + 13); p.bias[2] = F(base + 14);
  p.cin = cin; p.hid = hid; p.cout = cout;
  p.lincin = lincin; p.lincout = cout;
  p.linW = lincin ? F(base + 15) : nullptr;
  p.linb = lincin ? F(base + 16) : nullptr;
  return p;
}

void pack(const Ctx& c, const float* src, int sr, int sc, float* dst, int dr, int dc) {
  k_pack<<<(dr * dc + 255) / 256, 256, 0, c.st>>>(src, sr, sc, dst, dr, dc);
}

void pack_gnn(const Ctx& c, const GnnP& p, int s0) {
  pack(c, p.Wr[0], p.cin, p.hid, c.slot(s0 + 0), 64, 64);
  pack(c, p.Wx[0], p.cin, p.hid, c.slot(s0 + 1), 64, 64);
  pack(c, p.Wr[1], p.hid, p.hid, c.slot(s0 + 2), 64, 64);
  pack(c, p.Wx[1], p.hid, p.hid, c.slot(s0 + 3), 64, 64);
  pack(c, p.Wr[2], p.hid, p.cout, c.slot(s0 + 4), 64, 64);
  pack(c, p.Wx[2], p.hid, p.cout, c.slot(s0 + 5), 64, 64);
  if (p.lincin) pack(c, p.linW, p.lincin, p.lincout, c.slot(s0 + 6), 128, 64);
}

void gemm(const Ctx& c, const float* A, int lda, const float* B, int ldb,
          float* C, int ldc, int M, int N, int K, const float* addm,
          const float* bias, int biasN, const float* rs, int relu) {
  int tiles = (M / 16) * (N / 16);
  int blocks = (tiles + 7) / 8;
  k_gemm_wmma<<<blocks, 256, 0, c.st>>>(A, lda, B, ldb, C, ldc, M, N, K, addm,
                                        bias, biasN, rs, relu);
}

// one DenseSAGE layer + relu + BN (training-mode batch stats)
void run_sage(const Ctx& c, const float* X, const float* ADJ, int ldadj,
              const float* invdeg, int Mpad, int Kadj, int Nreal,
              const float* WrS, const float* WxS, const float* bias, int coutN,
              const float* g, const float* b, float* OUT) {
  float* AGG = c.xb(3);
  float* TMP = c.xb(4);
  float* STAT = c.W + OFF_STAT;
  gemm(c, ADJ, ldadj, X, 64, AGG, 64, Mpad, 64, Kadj, nullptr, nullptr, 0, invdeg, 0);
  gemm(c, AGG, 64, WrS, 64, TMP, 64, Mpad, 64, 64, nullptr, nullptr, 0, nullptr, 0);
  gemm(c, X, 64, WxS, 64, OUT, 64, Mpad, 64, 64, TMP, bias, coutN, nullptr, 1);
  k_bn_stats<<<64, 256, 0, c.st>>>(OUT, 64, Nreal, STAT);
  k_bn_apply<<<(Mpad * 64 + 255) / 256, 256, 0, c.st>>>(OUT, 64, Mpad, STAT, g, b, coutN);
}

// full GNN module: 3 sage layers -> concat -> optional lin+relu (into OUT)
void run_gnn(const Ctx& c, const GnnP& p, int s0, const float* X,
             const float* ADJ, int ldadj, const float* invdeg, int Mpad,
             int Kadj, int Nreal, float* CAT, float* OUT) {
  float* X1 = c.xb(0);
  float* X2 = c.xb(1);
  float* X3 = c.xb(2);
  run_sage(c, X, ADJ, ldadj, invdeg, Mpad, Kadj, Nreal, c.slot(s0 + 0),
           c.slot(s0 + 1), p.bias[0], p.hid, p.g[0], p.b[0], X1);
  run_sage(c, X1, ADJ, ldadj, invdeg, Mpad, Kadj, Nreal, c.slot(s0 + 2),
           c.slot(s0 + 3), p.bias[1], p.hid, p.g[1], p.b[1], X2);
  run_sage(c, X2, ADJ, ldadj, invdeg, Mpad, Kadj, Nreal, c.slot(s0 + 4),
           c.slot(s0 + 5), p.bias[2], p.cout, p.g[2], p.b[2], X3);
  k_copycol<<<(Mpad * p.hid + 255) / 256, 256, 0, c.st>>>(X1, 64, Mpad, p.hid, CAT, 128, 0);
  k_copycol<<<(Mpad * p.hid + 255) / 256, 256, 0, c.st>>>(X2, 64, Mpad, p.hid, CAT, 128, p.hid);
  k_copycol<<<(Mpad * p.cout + 255) / 256, 256, 0, c.st>>>(X3, 64, Mpad, p.cout, CAT, 128, 2 * p.hid);
  if (p.lincin)
    gemm(c, CAT, 128, c.slot(s0 + 6), 64, OUT, 64, Mpad, 64, 128, nullptr,
         p.linb, p.lincout, nullptr, 1);
}

}  // namespace

extern "C" void kernel_launch(void* const* d_in, const int* in_sizes, int n_in,
                              void* d_out, int out_size, void* d_ws,
                              size_t ws_size, hipStream_t stream) {
  // JAX pytree (sorted-key) leaf order: params first, then x.
  auto F = [&](int i) { return (const float*)d_in[i]; };
  const float* conv1W = F(0);
  const float* conv1b = F(1);
  const float* conv2W = F(2);
  const float* conv2b = F(3);
  const float* fcW = F(4);
  const float* fcb = F(5);
  GnnP g1e = mk_gnn(d_in, 6, 50, 24, 2, 0);
  GnnP g1p = mk_gnn(d_in, 21, 50, 32, 7, 71);
  GnnP g2e = mk_gnn(d_in, 38, 50, 24, 2, 0);
  GnnP g2p = mk_gnn(d_in, 53, 50, 32, 2, 66);
  GnnP g3e = mk_gnn(d_in, 70, 50, 32, 32, 0);
  const float* l1W = F(85);
  const float* l1b = F(86);
  const float* l2W = F(87);
  const float* l2b = F(88);
  const float* x = F(89);
  float* out = (float*)d_out;

  Ctx c{(float*)d_ws, stream};
  float* ADJ1 = c.W + OFF_ADJ1;
  float* H = c.W + OFF_H;
  float* SQ = c.W + OFF_SQ;
  float* IDG1 = c.W + OFF_IDG1;
  float* IDG2 = c.W + OFF_IDG2;
  float* IDG3 = c.W + OFF_IDG3;
  unsigned int* HIST = (unsigned int*)(c.W + OFF_HIST);
  float* SCAL = c.W + OFF_SCAL;
  float* CATA = c.W + OFF_CATA;
  float* CATB = c.W + OFF_CATB;
  float* S0 = c.xb(5);
  float* S2 = c.xb(6);
  float* T = c.xb(7);
  float* XG2 = c.xb(8);
  float* ADJ2 = c.xb(9);
  float* XG3 = c.xb(10);
  float* ADJ3 = c.xb(11);

  // 0) zero everything downstream of the adjacency slab (padding invariants,
  //    histogram, loss accumulators). Cheap: ~21 MB at HBM rate.
  k_zero<<<2048, 256, 0, stream>>>(H, (long long)(OFF_END - OFF_H));

  // 1) pack all GNN weights into zero-padded 64x64 / 128x64 WMMA slots
  pack_gnn(c, g1p, 0);
  pack_gnn(c, g1e, 7);
  pack_gnn(c, g2p, 14);
  pack_gnn(c, g2e, 21);
  pack_gnn(c, g3e, 28);

  // 2) fused CNN + FC per image -> H[4096,64] (padded), sq = rowwise |H|^2
  k_cnn<<<4096, 256, 0, stream>>>(x, conv1W, conv1b, conv2W, conv2b, fcW, fcb, H, SQ);

  // 3) D2 = |Hi - Hj|^2 via WMMA Gram; threshold search; adjacency; degrees
  k_gram_d2<<<(256 * 256 + 7) / 8, 256, 0, stream>>>(H, SQ, ADJ1, 4096);
  k_edge_hist<<<2048, 256, 0, stream>>>(ADJ1, 4096, HIST);
  k_pick_thr<<<1, 32, 0, stream>>>(HIST, SCAL, 409.6f);
  k_adj_build<<<4096, 256, 0, stream>>>(ADJ1, 4096, SCAL);
  k_invdeg<<<4096, 256, 0, stream>>>(ADJ1, 4096, 4096, IDG1);

  // 4) stage 1 (N=4096): pool + embed GNNs, DiffPool
  run_gnn(c, g1p, 0, H, ADJ1, 4096, IDG1, 4096, 4096, 4096, CATA, S0);
  run_gnn(c, g1e, 7, H, ADJ1, 4096, IDG1, 4096, 4096, 4096, CATB, nullptr);
  k_softmax_rows<<<16, 256, 0, stream>>>(S0, 64, 4096, 7);
  k_stx<<<4, 256, 0, stream>>>(S0, 64, CATB, 128, 4096, XG2);       // xn = S^T X
  gemm(c, ADJ1, 4096, S0, 64, T, 64, 4096, 64, 4096, nullptr, nullptr, 0, nullptr, 0);
  k_stx<<<4, 256, 0, stream>>>(S0, 64, T, 64, 4096, ADJ2);          // adjn = S^T A S
  k_link<<<2048, 256, 0, stream>>>(ADJ1, 4096, S0, 64, 4096, SCAL + 1);
  k_ent<<<16, 256, 0, stream>>>(S0, 64, 4096, 7, SCAL + 3);

  // 5) stage 2 (N=7, padded to 16): same kernels, tiny sizes
  k_invdeg<<<16, 256, 0, stream>>>(ADJ2, 64, 64, IDG2);
  run_gnn(c, g2p, 14, XG2, ADJ2, 64, IDG2, 16, 64, 7, CATA, S2);
  run_gnn(c, g2e, 21, XG2, ADJ2, 64, IDG2, 16, 64, 7, CATB, nullptr);
  k_softmax_rows<<<1, 32, 0, stream>>>(S2, 64, 7, 2);
  k_stx<<<4, 256, 0, stream>>>(S2, 64, CATB, 128, 7, XG3);
  gemm(c, ADJ2, 64, S2, 64, T, 64, 16, 64, 64, nullptr, nullptr, 0, nullptr, 0);
  k_stx<<<4, 256, 0, stream>>>(S2, 64, T, 64, 7, ADJ3);
  k_link<<<1, 256, 0, stream>>>(ADJ2, 64, S2, 64, 7, SCAL + 2);
  k_ent<<<1, 32, 0, stream>>>(S2, 64, 7, 2, SCAL + 4);

  // 6) stage 3 (N=2): gnn3_embed -> concat(96) -> mean -> lin1 -> lin2 -> lsm
  k_invdeg<<<16, 256, 0, stream>>>(ADJ3, 64, 64, IDG3);
  run_gnn(c, g3e, 28, XG3, ADJ3, 64, IDG3, 16, 64, 2, CATA, nullptr);
  k_final<<<1, 128, 0, stream>>>(CATA, 128, 2, l1W, l1b, l2W, l2b, SCAL, out);

  (void)in_sizes; (void)n_in; (void)out_size; (void)ws_size;
}